// RelationalEmbedding_20615843021266
// MI455X (gfx1250) — compile-verified
//
#include <hip/hip_runtime.h>
#include <hip/hip_bf16.h>

#define NN   768
#define UU   64
#define KK   16
#define DIN  144
#define DPAD 148   // padded LDS stride (kills 8*144 % 64 == 0 bank conflicts)
#define BBATCH 2
#define ITILES (NN/8)   // 96
#define JTILES (NN/16)  // 48

typedef __attribute__((ext_vector_type(2))) float v2f;
typedef __attribute__((ext_vector_type(8))) float v8f;

__device__ __forceinline__ float leaky01(float x) { return x > 0.f ? x : 0.01f * x; }

// ---------------------------------------------------------------- deg ----
__global__ void deg_kernel(const int* __restrict__ mask, float* __restrict__ rdeg) {
    int i = blockIdx.x * blockDim.x + threadIdx.x;
    if (i < NN) {
        int s = 0;
        for (int j = 0; j < NN; ++j) s += mask[(size_t)i * NN + j];
        rdeg[i] = 1.0f / (float)s;
    }
}

// ------------------------------------------------------- node projections ----
// A[b,n,c] = sum_u seq[b,n,u] * W1[u,c]        (e_i part)
// Bv[b,n,c] = sum_u seq[b,n,u] * W1[64+u,c]    (e_j part)
__global__ void proj_kernel(const float* __restrict__ seq, const float* __restrict__ W1,
                            float* __restrict__ A, float* __restrict__ Bv) {
    int idx = blockIdx.x * blockDim.x + threadIdx.x;  // over B*N*DIN
    if (idx >= BBATCH * NN * DIN) return;
    int c  = idx % DIN;
    int bn = idx / DIN;
    const float* srow = seq + (size_t)bn * UU;
    float s1 = 0.f, s2 = 0.f;
    for (int u = 0; u < UU; ++u) {
        float e = srow[u];
        s1 += e * W1[(size_t)u * DIN + c];
        s2 += e * W1[(size_t)(UU + u) * DIN + c];
    }
    A[idx]  = s1;
    Bv[idx] = s2;
}

// ---------------------------------------------------------- edge kernel ----
// grid: (JTILES, ITILES); block: 256 threads = 8 waves (wave32).
// Wave w handles i = i0+w and j = j0..j0+15 (16 edge pairs), both batches.
// E[i,j,c] = enc[i,j,:] @ W1[128:144, c] via V_WMMA_F32_16X16X4_F32.
__launch_bounds__(256)
__global__ void edge_kernel(const float* __restrict__ enc,
                            const int*   __restrict__ mask,
                            const float* __restrict__ A,
                            const float* __restrict__ Bv,
                            const float* __restrict__ W1,   // (144,144); rows 128..143 used
                            const float* __restrict__ b1,   // (144,)
                            const float* __restrict__ W2,   // (144,)
                            const float* __restrict__ b2,   // (1,)
                            const float* __restrict__ rdeg, // (N,)
                            float* __restrict__ partial)    // [B][ITILES][N]
{
    __shared__ float sW1h[KK][DPAD];          // W1 rows 128..143
    __shared__ float sA [BBATCH][8][DPAD];
    __shared__ float sBv[BBATCH][16][DPAD];
    __shared__ float sb1[DIN];
    __shared__ float sW2[DIN];
    __shared__ float sPart[BBATCH][8][16];

    const int tid   = threadIdx.x;
    const int j0    = blockIdx.x * 16;
    const int itile = blockIdx.y;
    const int i0    = itile * 8;

    for (int idx = tid; idx < KK * DIN; idx += 256) {
        int k = idx / DIN, c = idx % DIN;
        sW1h[k][c] = W1[(size_t)(2 * UU + k) * DIN + c];
    }
    for (int idx = tid; idx < BBATCH * 8 * DIN; idx += 256) {
        int c = idx % DIN, r = (idx / DIN) % 8, b = idx / (8 * DIN);
        sA[b][r][c] = A[((size_t)b * NN + i0 + r) * DIN + c];
    }
    for (int idx = tid; idx < BBATCH * 16 * DIN; idx += 256) {
        int c = idx % DIN, r = (idx / DIN) % 16, b = idx / (16 * DIN);
        sBv[b][r][c] = Bv[((size_t)b * NN + j0 + r) * DIN + c];
    }
    for (int idx = tid; idx < DIN; idx += 256) { sb1[idx] = b1[idx]; sW2[idx] = W2[idx]; }
    __syncthreads();

    const int wave = tid >> 5;
    const int lane = tid & 31;
    const int lm   = lane & 15;
    const int hi   = lane >> 4;       // 0: rows/K low half; 1: rows/K high half
    const int koff = hi * 2;          // f32 16x16x4: low lanes K0,K1 ; high lanes K2,K3
    const int i    = i0 + wave;
    const float rd  = rdeg[i];
    const float b2v = b2[0];

    // A-matrix fragments (enc row for this pair) for the 4 K-steps (K=16 total)
    const float* erow = enc + ((size_t)i * NN + (j0 + lm)) * KK;
    v2f af[4];
#pragma unroll
    for (int s = 0; s < 4; ++s) {
        af[s].x = erow[4 * s + koff];
        af[s].y = erow[4 * s + koff + 1];
    }

    float acc0[8], acc1[8];
#pragma unroll
    for (int r = 0; r < 8; ++r) { acc0[r] = 0.f; acc1[r] = 0.f; }

    for (int ct = 0; ct < 9; ++ct) {   // 9 channel tiles of 16 -> DIN=144
        const int c = ct * 16 + lm;

        // issue all B-fragment LDS loads up front so waits before the wmma
        // accumulator chain are incremental (dscnt 3,2,1,0) instead of full
        v2f bf[4];
#pragma unroll
        for (int s = 0; s < 4; ++s) {
            bf[s].x = sW1h[4 * s + koff][c];
            bf[s].y = sW1h[4 * s + koff + 1][c];
        }

        v8f d = {0.f, 0.f, 0.f, 0.f, 0.f, 0.f, 0.f, 0.f};
#pragma unroll
        for (int s = 0; s < 4; ++s) {
            d = __builtin_amdgcn_wmma_f32_16x16x4_f32(false, af[s], false, bf[s],
                                                      (short)0, d, false, false);
        }

        const float w2c   = sW2[c];
        const float aval0 = sA[0][wave][c] + sb1[c];
        const float aval1 = sA[1][wave][c] + sb1[c];
#pragma unroll
        for (int r = 0; r < 8; ++r) {
            const int row = r + 8 * hi;       // pair index -> j = j0 + row
            const float e = d[r];
            float t0 = e + aval0 + sBv[0][row][c];
            float t1 = e + aval1 + sBv[1][row][c];
            t0 = leaky01(t0);
            t1 = leaky01(t1);
            acc0[r] += t0 * w2c;
            acc1[r] += t1 * w2c;
        }
    }

    // sum over channels: butterfly across the 16 lanes of each half-wave
#pragma unroll
    for (int off = 1; off < 16; off <<= 1) {
#pragma unroll
        for (int r = 0; r < 8; ++r) {
            acc0[r] += __shfl_xor(acc0[r], off, 16);
            acc1[r] += __shfl_xor(acc1[r], off, 16);
        }
    }

    if (lm == 0) {   // lanes 0 and 16 hold rows 0..7 and 8..15 respectively
#pragma unroll
        for (int r = 0; r < 8; ++r) {
            const int row = r + 8 * hi;
            const int m   = mask[(size_t)i * NN + j0 + row];
            const float fm = (float)m * rd;
            sPart[0][wave][row] = leaky01(acc0[r] + b2v) * fm;
            sPart[1][wave][row] = leaky01(acc1[r] + b2v) * fm;
        }
    }
    __syncthreads();

    if (tid < BBATCH * 16) {   // reduce the 8 waves (8 i's) of this block
        const int b = tid >> 4, row = tid & 15;
        float s = 0.f;
#pragma unroll
        for (int w = 0; w < 8; ++w) s += sPart[b][w][row];
        partial[((size_t)b * ITILES + itile) * NN + j0 + row] = s;
    }
}

// ------------------------------------------------------------- finish ----
// s[b,j] = sum over i-tiles of partial; seq_out[b,j,:] = s * seq_in[b,j,:]
__global__ void finish_kernel(const float* __restrict__ partial,
                              const float* __restrict__ seq_in,
                              float* __restrict__ seq_out) {
    const int bj = blockIdx.x;
    const int b = bj / NN, j = bj % NN;
    __shared__ float red[128];
    const int t = threadIdx.x;
    float v = 0.f;
    if (t < ITILES) v = partial[((size_t)b * ITILES + t) * NN + j];
    red[t] = v;
    __syncthreads();
    for (int off = 64; off > 0; off >>= 1) {
        if (t < off) red[t] += red[t + off];
        __syncthreads();
    }
    const float s = red[0];
    if (t < UU) {
        const size_t base = ((size_t)b * NN + j) * UU + t;
        seq_out[base] = s * seq_in[base];
    }
}

extern "C" void kernel_launch(void* const* d_in, const int* in_sizes, int n_in,
                              void* d_out, int out_size, void* d_ws, size_t ws_size,
                              hipStream_t stream) {
    const float* seq  = (const float*)d_in[0];
    const float* enc  = (const float*)d_in[1];
    const int*   mask = (const int*)d_in[2];
    const float* W1_1 = (const float*)d_in[3];
    const float* b1_1 = (const float*)d_in[4];
    const float* W1_2 = (const float*)d_in[5];
    const float* b1_2 = (const float*)d_in[6];
    const float* W2_1 = (const float*)d_in[7];
    const float* b2_1 = (const float*)d_in[8];
    const float* W2_2 = (const float*)d_in[9];
    const float* b2_2 = (const float*)d_in[10];
    float* out = (float*)d_out;

    // workspace layout (floats): ~2.8 MB total
    float* ws      = (float*)d_ws;
    float* A       = ws;                          // B*N*DIN = 221184
    float* Bv      = A    + BBATCH * NN * DIN;    // 221184
    float* seq1    = Bv   + BBATCH * NN * DIN;    // B*N*U   =  98304
    float* rdeg    = seq1 + BBATCH * NN * UU;     // N       =    768
    float* partial = rdeg + NN;                   // B*ITILES*N = 147456

    dim3 gEdge(JTILES, ITILES);
    const int projN = BBATCH * NN * DIN;

    deg_kernel<<<(NN + 255) / 256, 256, 0, stream>>>(mask, rdeg);

    // hop 1: params ((W1_1,b1_1),(W1_2,b1_2))
    proj_kernel<<<(projN + 255) / 256, 256, 0, stream>>>(seq, W1_1, A, Bv);
    edge_kernel<<<gEdge, 256, 0, stream>>>(enc, mask, A, Bv, W1_1, b1_1, W1_2, b1_2,
                                           rdeg, partial);
    finish_kernel<<<BBATCH * NN, 128, 0, stream>>>(partial, seq, seq1);

    // hop 2: params ((W2_1,b2_1),(W2_2,b2_2))
    proj_kernel<<<(projN + 255) / 256, 256, 0, stream>>>(seq1, W2_1, A, Bv);
    edge_kernel<<<gEdge, 256, 0, stream>>>(enc, mask, A, Bv, W2_1, b2_1, W2_2, b2_2,
                                           rdeg, partial);
    finish_kernel<<<BBATCH * NN, 128, 0, stream>>>(partial, seq1, out);
}